// GraphConvBase_86775519249037
// MI455X (gfx1250) — compile-verified
//
#include <hip/hip_runtime.h>
#include <hip/hip_bf16.h>

#define D 64

typedef float v2f __attribute__((ext_vector_type(2)));
typedef float v8f __attribute__((ext_vector_type(8)));

// ---------------------------------------------------------------- zero
__global__ void zero_f32_kernel(float* __restrict__ p, int n) {
    int i = blockIdx.x * blockDim.x + threadIdx.x;
    if (i < n) p[i] = 0.0f;
}

// ---------------------------------------------------------------- scatter-add (gather x[src], add into agg[dst])
__global__ void scatter_add_kernel(const float* __restrict__ x,
                                   const long long* __restrict__ src,
                                   const long long* __restrict__ dst,
                                   float* __restrict__ agg, int nEdges) {
    int t = blockIdx.x * blockDim.x + threadIdx.x;
    int total = nEdges * 16;               // 16 float4 chunks per edge row of 64
    if (t >= total) return;
    int e = t >> 4;
    int f = (t & 15) << 2;
    int s = (int)src[e];
    int d = (int)dst[e];
    const float4 v = *(const float4*)(x + (size_t)s * D + f);
    float* q = agg + (size_t)d * D + f;
    atomicAdd(q + 0, v.x);
    atomicAdd(q + 1, v.y);
    atomicAdd(q + 2, v.z);
    atomicAdd(q + 3, v.w);
}

// ---------------------------------------------------------------- fused GraphConv GEMM via V_WMMA_F32_16X16X4_F32
// Out[n, o] = sum_k A[n,k]*Wrel[o,k] + sum_k X[n,k]*Wroot[o,k] + bias[o]   (optional ReLU)
// 4 waves / block, each wave computes one 16-row x 64-col output tile.
__launch_bounds__(128)
__global__ void graphconv_gemm_wmma(const float* __restrict__ A,
                                    const float* __restrict__ X,
                                    const float* __restrict__ Wrel,
                                    const float* __restrict__ Wroot,
                                    const float* __restrict__ bias,
                                    float* __restrict__ Out,
                                    int nRows, int nTiles, int doRelu) {
    __shared__ float sWrel[D * 65];   // row-padded to 65 to avoid LDS bank conflicts
    __shared__ float sWroot[D * 65];
    __shared__ float sBias[D];

    // Cooperative stage of both 64x64 weight matrices + bias into LDS (all 128 threads).
    for (int i = threadIdx.x; i < D * D; i += 128) {
        int r = i >> 6, c = i & 63;
        sWrel[r * 65 + c]  = Wrel[i];
        sWroot[r * 65 + c] = Wroot[i];
    }
    if (threadIdx.x < D) sBias[threadIdx.x] = bias[threadIdx.x];
    __syncthreads();

    int wave = threadIdx.x >> 5;
    int lane = threadIdx.x & 31;
    int tile = blockIdx.x * 4 + wave;
    if (tile >= nTiles) return;            // wave-uniform: EXEC stays all-1s for WMMA

    int l16 = lane & 15;
    int hi  = (lane >> 4) & 1;             // lanes 16-31 carry K+2/K+3 (A) / M+8 (C)
    int row = tile * 16 + l16;             // A-matrix M index owned by this lane
    if (row >= nRows) row = nRows - 1;     // clamp loads (stores guarded below)
    const float* arow = A + (size_t)row * D;
    const float* xrow = X + (size_t)row * D;

    v8f z = {0.f, 0.f, 0.f, 0.f, 0.f, 0.f, 0.f, 0.f};
    v8f acc[4];
    acc[0] = z; acc[1] = z; acc[2] = z; acc[3] = z;

    #pragma unroll
    for (int kb = 0; kb < 16; ++kb) {      // K = 64 in steps of 4
        int k0 = kb * 4 + hi * 2;
        v2f aA, aX;
        aA.x = arow[k0]; aA.y = arow[k0 + 1];
        aX.x = xrow[k0]; aX.y = xrow[k0 + 1];
        #pragma unroll
        for (int cb = 0; cb < 4; ++cb) {   // 4 column tiles of 16
            int o = cb * 16 + l16;         // output-feature index (B column)
            v2f bR, bG;
            bR.x = sWrel[o * 65 + k0];  bR.y = sWrel[o * 65 + k0 + 1];
            bG.x = sWroot[o * 65 + k0]; bG.y = sWroot[o * 65 + k0 + 1];
            acc[cb] = __builtin_amdgcn_wmma_f32_16x16x4_f32(
                false, aA, false, bR, (short)0, acc[cb], false, false);
            acc[cb] = __builtin_amdgcn_wmma_f32_16x16x4_f32(
                false, aX, false, bG, (short)0, acc[cb], false, false);
        }
    }

    // Epilogue: C/D layout -> VGPR j holds M=j (lanes 0-15) or M=8+j (lanes 16-31), N=lane%16
    #pragma unroll
    for (int cb = 0; cb < 4; ++cb) {
        int col = cb * 16 + l16;
        float b = sBias[col];
        #pragma unroll
        for (int j = 0; j < 8; ++j) {
            int orow = tile * 16 + j + hi * 8;
            float v = acc[cb][j] + b;
            if (doRelu) v = v > 0.f ? v : 0.f;
            if (orow < nRows) Out[(size_t)orow * D + col] = v;
        }
    }
}

// ---------------------------------------------------------------- global mean pool (sum + count via atomics)
__global__ void pool_kernel(const float* __restrict__ emb,
                            const long long* __restrict__ batch,
                            float* __restrict__ pooled,
                            float* __restrict__ cnt, int nNodes) {
    int t = blockIdx.x * blockDim.x + threadIdx.x;
    if (t >= nNodes * D) return;
    int node = t >> 6;
    int f = t & 63;
    int g = (int)batch[node];
    float v = emb[(size_t)node * D + f];
    v = v > 0.f ? v : 0.f;                 // relu before pooling
    atomicAdd(pooled + g * D + f, v);
    if (f == 0) atomicAdd(cnt + g, 1.0f);
}

// ---------------------------------------------------------------- post_mp: (pooled @ W4^T + b4) @ W5^T + b5
__launch_bounds__(64)
__global__ void mlp_kernel(const float* __restrict__ pooled, const float* __restrict__ cnt,
                           const float* __restrict__ W4, const float* __restrict__ b4,
                           const float* __restrict__ W5, const float* __restrict__ b5,
                           float* __restrict__ out) {
    __shared__ float sp[D];
    __shared__ float sh[D];
    int g = blockIdx.x, t = threadIdx.x;
    float c = cnt[g];
    c = c < 1.f ? 1.f : c;
    sp[t] = pooled[g * D + t] / c;
    __syncthreads();
    float a = b4[t];
    #pragma unroll 8
    for (int k = 0; k < D; ++k) a += sp[k] * W4[t * D + k];
    sh[t] = a;                              // no activation between the two linears
    __syncthreads();
    if (t < 32) {
        float o = b5[t];
        #pragma unroll 8
        for (int k = 0; k < D; ++k) o += sh[k] * W5[t * D + k];
        out[g * 32 + t] = o;
    }
}

// ---------------------------------------------------------------- launch
extern "C" void kernel_launch(void* const* d_in, const int* in_sizes, int n_in,
                              void* d_out, int out_size, void* d_ws, size_t ws_size,
                              hipStream_t stream) {
    const float*     x     = (const float*)d_in[0];
    const long long* eidx  = (const long long*)d_in[1];   // int64 (2, E)
    const long long* batch = (const long long*)d_in[2];   // int64 (N,)
    const float* W1r = (const float*)d_in[3];
    const float* b1  = (const float*)d_in[4];
    const float* W1g = (const float*)d_in[5];
    const float* W2r = (const float*)d_in[6];
    const float* b2  = (const float*)d_in[7];
    const float* W2g = (const float*)d_in[8];
    const float* W3r = (const float*)d_in[9];
    const float* b3  = (const float*)d_in[10];
    const float* W3g = (const float*)d_in[11];
    const float* W4  = (const float*)d_in[12];
    const float* b4  = (const float*)d_in[13];
    const float* W5  = (const float*)d_in[14];
    const float* b5  = (const float*)d_in[15];

    const int nNodes  = in_sizes[0] / D;                  // 50000
    const int nEdges  = in_sizes[1] / 2;                  // 800000
    const int nGraphs = (out_size - nNodes * D) / 32;     // 64

    // workspace carve-out
    float* agg    = (float*)d_ws;                         // N*64
    float* h1     = agg + (size_t)nNodes * D;             // N*64
    float* h2     = h1  + (size_t)nNodes * D;             // N*64
    float* pooled = h2  + (size_t)nNodes * D;             // G*64
    float* cnt    = pooled + (size_t)nGraphs * D;         // G
    float* emb    = (float*)d_out;                        // output 0: N*64
    float* outp   = emb + (size_t)nNodes * D;             // output 1: G*32

    const long long* src = eidx;
    const long long* dst = eidx + nEdges;

    const int nAgg   = nNodes * D;
    const int zGrid  = (nAgg + 255) / 256;
    const int sGrid  = (nEdges * 16 + 255) / 256;
    const int nTiles = (nNodes + 15) / 16;
    const int gGrid  = (nTiles + 3) / 4;

    // ---- layer 1: h1 = relu(GraphConv(x))
    zero_f32_kernel<<<zGrid, 256, 0, stream>>>(agg, nAgg);
    scatter_add_kernel<<<sGrid, 256, 0, stream>>>(x, src, dst, agg, nEdges);
    graphconv_gemm_wmma<<<gGrid, 128, 0, stream>>>(agg, x, W1r, W1g, b1, h1, nNodes, nTiles, 1);

    // ---- layer 2: h2 = relu(GraphConv(h1))
    zero_f32_kernel<<<zGrid, 256, 0, stream>>>(agg, nAgg);
    scatter_add_kernel<<<sGrid, 256, 0, stream>>>(h1, src, dst, agg, nEdges);
    graphconv_gemm_wmma<<<gGrid, 128, 0, stream>>>(agg, h1, W2r, W2g, b2, h2, nNodes, nTiles, 1);

    // ---- layer 3: emb = GraphConv(h2)   (no relu; relu applied at pooling)
    zero_f32_kernel<<<zGrid, 256, 0, stream>>>(agg, nAgg);
    scatter_add_kernel<<<sGrid, 256, 0, stream>>>(h2, src, dst, agg, nEdges);
    graphconv_gemm_wmma<<<gGrid, 128, 0, stream>>>(agg, h2, W3r, W3g, b3, emb, nNodes, nTiles, 0);

    // ---- global mean pool of relu(emb)
    const int nPool = nGraphs * D + nGraphs;              // pooled + cnt are contiguous
    zero_f32_kernel<<<(nPool + 255) / 256, 256, 0, stream>>>(pooled, nPool);
    pool_kernel<<<(nNodes * D + 255) / 256, 256, 0, stream>>>(emb, batch, pooled, cnt, nNodes);

    // ---- post_mp
    mlp_kernel<<<nGraphs, 64, 0, stream>>>(pooled, cnt, W4, b4, W5, b5, outp);
}